// URandKnnPool_33036888441077
// MI455X (gfx1250) — compile-verified
//
#include <hip/hip_runtime.h>
#include <hip/hip_bf16.h>

// Problem constants (match reference)
#define BATCHES 8
#define PTS     4096
#define KM1     31          // K-1 neighbors after self-loop filtering
#define NTOT    (BATCHES*PTS)

// Output sections (floats), concatenated in reference return order:
// node_coord_dst [N*3], edge_src [N*31], edge_dst [N*31], degree [N],
// batch_dst [N], node_dst_idx [N]
#define OFF_COORD 0
#define OFF_ESRC  (NTOT*3)
#define OFF_EDST  (OFF_ESRC + NTOT*KM1)
#define OFF_DEG   (OFF_EDST + NTOT*KM1)
#define OFF_BATCH (OFF_DEG + NTOT)
#define OFF_IDX   (OFF_BATCH + NTOT)

typedef float v2f __attribute__((ext_vector_type(2)));
typedef float v8f __attribute__((ext_vector_type(8)));

// ---------------------------------------------------------------------------
// Kernel 1: all trivial outputs (coord copy, edge_dst, degree=31, batch, idx)
// ---------------------------------------------------------------------------
__global__ __launch_bounds__(256) void urkp_misc_kernel(
    const float* __restrict__ coord, float* __restrict__ out)
{
    int i = blockIdx.x * blockDim.x + threadIdx.x;
    if (i >= NTOT) return;
    out[OFF_COORD + 3*i + 0] = coord[3*i + 0];
    out[OFF_COORD + 3*i + 1] = coord[3*i + 1];
    out[OFF_COORD + 3*i + 2] = coord[3*i + 2];
    float fi = (float)i;
    for (int k = 0; k < KM1; ++k)
        out[OFF_EDST + (size_t)i*KM1 + k] = fi;     // edge_dst = repeat(i, 31)
    out[OFF_DEG   + i] = 31.0f;                      // exact: every dst has 31 edges
    out[OFF_BATCH + i] = (float)(i >> 12);           // i / 4096
    out[OFF_IDX   + i] = fi;                         // identity subsample
}

// ---------------------------------------------------------------------------
// Kernel 2: batched KNN edge_src via WMMA f32 16x16x4 distance tiles.
//   rank value r_ij = |x_j|^2 - 2 x_i.x_j   (row-constant |x_i|^2 dropped)
//   A row i = [x,y,z,1], B col j = [-2x', -2y', -2z', |x'|^2]  -> K=4 fp32 WMMA
// Block: 256 threads = 8 waves; wave owns 16 query rows; 64-col chunks.
// B-tile coordinate loads are software-pipelined one chunk ahead so the
// global_load_b96 latency is hidden under the selection VALU work.
// ---------------------------------------------------------------------------
__global__ __launch_bounds__(256) void urkp_knn_kernel(
    const float* __restrict__ coord, float* __restrict__ out)
{
    // Per-wave 16x64 distance tile, padded to 65 dwords/row: read-back
    // addr = r*65 + c  ->  bank = (r + c) % 64, conflict-free across lanes.
    __shared__ float dtile[8][16][65];

    const int lane = threadIdx.x & 31;
    const int w    = threadIdx.x >> 5;
    const int m    = lane & 15;          // row within wave tile / col within B tile
    const bool hi  = lane >= 16;
    const int h    = lane >> 4;          // which half of the 64 columns this lane scans

    const int blocksPerBatch = PTS / 128;            // 32
    const int b    = blockIdx.x / blocksPerBatch;
    const int rb   = blockIdx.x % blocksPerBatch;
    const int l0   = rb * 128 + w * 16;              // wave's first local row
    const int base = b * PTS;                        // batch offset into point array

    // ---- A tile (16x4 f32): lanes 0-15 hold K0,K1; lanes 16-31 hold K2,K3 ----
    const float* rp = coord + (size_t)(base + l0 + m) * 3;
    const float rx = rp[0], ry = rp[1], rz = rp[2];
    v2f a;
    a.x = hi ? rz   : rx;     // K2 : K0
    a.y = hi ? 1.0f : ry;     // K3 : K1   (homogeneous 1 -> picks up |x_j|^2 from B)

    // ---- register-resident unsorted top-31 with tracked max (threshold) ----
    float bestv[KM1];
    int   besti[KM1];
#pragma unroll
    for (int t = 0; t < KM1; ++t) { bestv[t] = __builtin_inff(); besti[t] = 0; }
    float curmax  = __builtin_inff();
    int   maxslot = 0;

    const int il = l0 + m;               // local row this lane ranks for

    // ---- pipeline prologue: load chunk 0's B-tile coordinates ----
    float qx[4], qy[4], qz[4];
#pragma unroll
    for (int cc = 0; cc < 4; ++cc) {
        const float* qp = coord + (size_t)(base + cc * 16 + m) * 3;
        qx[cc] = qp[0]; qy[cc] = qp[1]; qz[cc] = qp[2];
    }

    for (int chunk = 0; chunk < PTS / 64; ++chunk) {
        const int jc0 = chunk * 64;

        // ---- 4 x WMMA: 16 rows x 64 cols of rank values (uses prefetched q) ----
#pragma unroll
        for (int cc = 0; cc < 4; ++cc) {
            const float sq = qx[cc]*qx[cc] + qy[cc]*qy[cc] + qz[cc]*qz[cc];
            v2f bb;
            bb.x = hi ? -2.0f*qz[cc] : -2.0f*qx[cc];   // K2 : K0
            bb.y = hi ?  sq          : -2.0f*qy[cc];   // K3 : K1
            v8f c = {0.f,0.f,0.f,0.f,0.f,0.f,0.f,0.f};
            c = __builtin_amdgcn_wmma_f32_16x16x4_f32(
                    false, a, false, bb, (short)0, c, false, false);
            // C layout: vgpr v, lanes 0-15 -> row v; lanes 16-31 -> row v+8
#pragma unroll
            for (int v = 0; v < 8; ++v) {
                const int row = hi ? v + 8 : v;
                dtile[w][row][cc * 16 + m] = c[v];
            }
        }

        // ---- issue next chunk's coordinate loads NOW; selection hides them ----
        const int njc = (jc0 + 64) & (PTS - 1);        // wraps to 0 on last iter
        __builtin_prefetch(coord + (size_t)(base + ((njc + 64) & (PTS - 1)) + lane) * 3, 0, 1);
#pragma unroll
        for (int cc = 0; cc < 4; ++cc) {
            const float* qp = coord + (size_t)(base + njc + cc * 16 + m) * 3;
            qx[cc] = qp[0]; qy[cc] = qp[1]; qz[cc] = qp[2];
        }

        // ---- selection: lane (m,h) scans 32 columns of row m ----
        for (int t = 0; t < 32; ++t) {
            const int cgl = h * 32 + t;
            const float v = dtile[w][m][cgl];
            const int jg  = jc0 + cgl;
            if (jg != il && v < curmax) {      // threshold: almost always 1 cmp
#pragma unroll
                for (int s = 0; s < KM1; ++s)
                    if (s == maxslot) { bestv[s] = v; besti[s] = jg; }
                float mx = -__builtin_inff(); int msl = 0;
#pragma unroll
                for (int s = 0; s < KM1; ++s)
                    if (bestv[s] > mx) { mx = bestv[s]; msl = s; }
                curmax = mx; maxslot = msl;
            }
        }
    }

    // ---- merge halves: lanes 0-15 absorb partner (lane+16) lists via shfl ----
#pragma unroll
    for (int t = 0; t < KM1; ++t) {
        const float pv = __shfl(bestv[t], (lane + 16) & 31, 32);
        const int   pi = __shfl(besti[t], (lane + 16) & 31, 32);
        if (!hi && pv < curmax) {
#pragma unroll
            for (int s = 0; s < KM1; ++s)
                if (s == maxslot) { bestv[s] = pv; besti[s] = pi; }
            float mx = -__builtin_inff(); int msl = 0;
#pragma unroll
            for (int s = 0; s < KM1; ++s)
                if (bestv[s] > mx) { mx = bestv[s]; msl = s; }
            curmax = mx; maxslot = msl;
        }
    }

    // ---- ascending-distance emission (matches top_k order) ----
    if (!hi) {
        const int gi = base + il;
        float* es = out + OFF_ESRC + (size_t)gi * KM1;
        for (int k = 0; k < KM1; ++k) {
            float mv = __builtin_inff(); int msl = 0;
#pragma unroll
            for (int s = 0; s < KM1; ++s)
                if (bestv[s] < mv) { mv = bestv[s]; msl = s; }
            int oi = 0;
#pragma unroll
            for (int s = 0; s < KM1; ++s)
                if (s == msl) { oi = besti[s]; bestv[s] = __builtin_inff(); }
            es[k] = (float)(base + oi);
        }
    }
}

// ---------------------------------------------------------------------------
extern "C" void kernel_launch(void* const* d_in, const int* in_sizes, int n_in,
                              void* d_out, int out_size, void* d_ws, size_t ws_size,
                              hipStream_t stream)
{
    (void)in_sizes; (void)n_in; (void)out_size; (void)d_ws; (void)ws_size;
    const float* coord = (const float*)d_in[0];   // node_coord_src [N,3] f32
    // d_in[1] (batch_src) is implied by the uniform contiguous batching.
    float* out = (float*)d_out;

    urkp_misc_kernel<<<(NTOT + 255) / 256, 256, 0, stream>>>(coord, out);
    urkp_knn_kernel<<<NTOT / 128, 256, 0, stream>>>(coord, out);
}